// ContactMapGather_13821204759244
// MI455X (gfx1250) — compile-verified
//
#include <hip/hip_runtime.h>

typedef float v2f __attribute__((ext_vector_type(2)));
typedef float v4f __attribute__((ext_vector_type(4)));
typedef float v8f __attribute__((ext_vector_type(8)));

#define Lr    512
#define Cc    128
#define NBAT  4
#define NBLK  32            // Lr / 16
#define NPAIR 528           // NBLK*(NBLK+1)/2 (block-upper-triangle incl. diagonal)
#define WPB   4             // waves per block
#define XSTR  132           // LDS row stride in dwords (conflict-free for A-frag b64 reads)

__global__ __launch_bounds__(128)
void pair_sym_proj_kernel(const float* __restrict__ feat,
                          const int*   __restrict__ flag,
                          const float* __restrict__ W,
                          const float* __restrict__ bias,
                          float*       __restrict__ out)
{
    __shared__ float wlds[32 * 2 * 32];          // [k-chunk][vgpr][lane] padded-W B-fragments
    __shared__ float xs[WPB][16 * XSTR];         // per-wave symmetrized 16x128 tile
    __shared__ float ob[WPB][32];                // per-wave 16 positions x 2 outputs

    const int tid   = threadIdx.x;
    const int lane  = tid & 31;
    const int wslot = tid >> 5;

    // ---- Stage zero-padded W as per-lane WMMA B-fragments (once per block) ----
    // B 4x16 layout (mirrors ISA A 16x4): V0 holds K=0 (lanes 0-15) / K=2 (lanes 16-31),
    // V1 holds K=1 / K=3. Column n = lane&15; only n<2 are real (W is 128x2).
    #pragma unroll
    for (int t = 0; t < 16; ++t) {
        int idx  = tid + t * 128;
        int k    = idx >> 6;
        int v    = (idx >> 5) & 1;
        int ln   = idx & 31;
        int n    = ln & 15;
        int koff = ((ln >> 4) << 1) + v;
        int c    = (k << 2) + koff;
        wlds[idx] = (n < 2) ? W[c * 2 + n] : 0.0f;
    }
    __syncthreads();

    // ---- Decode wave -> (batch, l, m-block) over block-upper-triangle ----
    const int wave = blockIdx.x * WPB + wslot;
    const int b    = wave / (NPAIR * 16);
    int r          = wave % (NPAIR * 16);
    const int li   = r & 15;
    int pair       = r >> 4;
    int lblk       = 0;
    while (pair >= NBLK - lblk) { pair -= (NBLK - lblk); ++lblk; }
    const int mblk = lblk + pair;              // mblk >= lblk
    const int l    = (lblk << 4) + li;
    const int m0   = mblk << 4;

    // ---- Fill symmetrized tile: xs[i][c] = 0.5*(relu(f[b,l,m0+i,c]) + relu(f[b,m0+i,l,c]))
    {
        const long baseA = (((long)b * Lr + l) * Lr + m0) * Cc;
        const long baseT = (((long)b * Lr + m0) * Lr + l) * Cc;
        const int  c0    = lane << 2;          // 4 channels per lane
        #pragma unroll 4
        for (int i = 0; i < 16; ++i) {
            v4f a = __builtin_nontemporal_load((const v4f*)(feat + baseA + (long)i * Cc + c0));
            v4f t = __builtin_nontemporal_load((const v4f*)(feat + baseT + (long)i * Lr * Cc + c0));
            v4f s;
            s.x = 0.5f * (fmaxf(a.x, 0.0f) + fmaxf(t.x, 0.0f));
            s.y = 0.5f * (fmaxf(a.y, 0.0f) + fmaxf(t.y, 0.0f));
            s.z = 0.5f * (fmaxf(a.z, 0.0f) + fmaxf(t.z, 0.0f));
            s.w = 0.5f * (fmaxf(a.w, 0.0f) + fmaxf(t.w, 0.0f));
            *(v4f*)&xs[wslot][i * XSTR + c0] = s;
        }
    }
    __syncthreads();

    // ---- Projection: D[16pos x 16col] = sum_k A[16x4] * B[4x16], K=128 in 32 chunks ----
    v8f acc = {0.f, 0.f, 0.f, 0.f, 0.f, 0.f, 0.f, 0.f};
    {
        const int row  = lane & 15;            // M (position within strip)
        const int koff = (lane >> 4) << 1;     // lanes 0-15: K+0/K+1, lanes 16-31: K+2/K+3
        const float* xw = &xs[wslot][row * XSTR + koff];
        #pragma unroll
        for (int k = 0; k < 32; ++k) {
            v2f a;  a.x  = xw[k * 4];             a.y  = xw[k * 4 + 1];
            v2f bb; bb.x = wlds[k * 64 + lane];   bb.y = wlds[k * 64 + 32 + lane];
            acc = __builtin_amdgcn_wmma_f32_16x16x4_f32(
                      false, a, false, bb, (short)0, acc, false, false);
        }
    }

    // ---- Gather the two real columns through LDS ----
    if ((lane & 15) < 2) {
        const int n     = lane & 15;
        const int mhalf = (lane >> 4) << 3;    // C/D layout: VGPR j -> M = j + 8*(lane>>4)
        #pragma unroll
        for (int j = 0; j < 8; ++j)
            ob[wslot][(mhalf + j) * 2 + n] = acc[j];
    }
    __syncthreads();

    // ---- Store direct + mirrored tiles with bias and flag mask ----
    {
        const int  M   = lane >> 1;
        const int  o   = lane & 1;
        const int  m   = m0 + M;
        const float v  = ob[wslot][lane] + bias[o];
        const long pd  = ((long)b * Lr + l) * Lr + m;   // (l, m)
        const long pm  = ((long)b * Lr + m) * Lr + l;   // (m, l)  (out is symmetric pre-mask)
        out[pd * 2 + o] = flag[pd] ? v : 0.0f;
        out[pm * 2 + o] = flag[pm] ? v : 0.0f;
    }
}

extern "C" void kernel_launch(void* const* d_in, const int* in_sizes, int n_in,
                              void* d_out, int out_size, void* d_ws, size_t ws_size,
                              hipStream_t stream) {
    const float* feat = (const float*)d_in[0];
    const int*   flag = (const int*)  d_in[1];
    const float* W    = (const float*)d_in[2];
    const float* bias = (const float*)d_in[3];
    float*       out  = (float*)d_out;

    const int total_waves = NBAT * NPAIR * 16;   // 33,792
    const int blocks      = total_waves / WPB;   // 8,448 (exact, no tail)
    hipLaunchKernelGGL(pair_sym_proj_kernel, dim3(blocks), dim3(128), 0, stream,
                       feat, flag, W, bias, out);
}